// GRU_DE_44968307589653
// MI455X (gfx1250) — compile-verified
//
#include <hip/hip_runtime.h>

// Autoregressive GRU decoder for gfx1250 (MI455X), fp32 via V_WMMA_F32_16X16X4_F32.
// B=32768 rows, H=32, T=128 sequential steps. One wave32 owns a 16-row tile for
// the whole recurrence; weights live in registers as WMMA B operands.
//
// - For t>=1 the GRU input x equals the previous hidden h, so the r/z gate
//   weights fuse: W_rz = W_ih[0:2H] + W_hh[0:2H]. Steady state = 64 WMMAs/step.
// - WMMA accumulator chains are seeded with inline-constant 0 (backend folds a
//   zeroinitializer C into src2=0), eliminating per-step bias-splat movs and
//   tied-register accumulator copies; biases are folded into the gate math.

typedef __attribute__((ext_vector_type(2))) float v2f;
typedef __attribute__((ext_vector_type(8))) float v8f;

#define GRU_H 32
#define TILE_ROWS 16
#define WAVES_PER_BLOCK 8
#define ROWS_PER_BLOCK (TILE_ROWS * WAVES_PER_BLOCK)
#define LDS_STRIDE 34  // padded row stride (floats): bank-conflict free, 8B aligned

__device__ __forceinline__ float fast_sigmoid(float x) {
  return __builtin_amdgcn_rcpf(1.0f + __expf(-x));
}
__device__ __forceinline__ float fast_tanh(float x) {
  float e = __expf(-2.0f * x);
  return (1.0f - e) * __builtin_amdgcn_rcpf(1.0f + e);
}

__device__ __forceinline__ v8f wmma4(v2f a, v2f b, v8f c) {
  // D = A(16x4 f32) * B(4x16 f32) + C(16x16 f32)
  return __builtin_amdgcn_wmma_f32_16x16x4_f32(
      /*neg_a=*/false, a, /*neg_b=*/false, b,
      /*c_mod=*/(short)0, c, /*reuse_a=*/false, /*reuse_b=*/false);
}

// LDS executes in-order within a wave; stop compiler reordering of cross-lane
// LDS traffic and conservatively drain DScnt.
__device__ __forceinline__ void wave_lds_sync() {
  __builtin_amdgcn_wave_barrier();
  asm volatile("s_wait_dscnt 0x0" ::: "memory");
  __builtin_amdgcn_wave_barrier();
}

__global__ __launch_bounds__(256) void GRU_DE_44968307589653_kernel(
    const float* __restrict__ in_data, const float* __restrict__ hidden,
    const float* __restrict__ W_ih, const float* __restrict__ W_hh,
    const float* __restrict__ b_ih, const float* __restrict__ b_hh,
    const float* __restrict__ W_out, const float* __restrict__ b_out,
    const int* __restrict__ num_steps, float* __restrict__ out, int B) {
  __shared__ float smem[WAVES_PER_BLOCK * TILE_ROWS * LDS_STRIDE];

  const int lane = threadIdx.x & 31;
  const int wid = threadIdx.x >> 5;
  const int row0 = blockIdx.x * ROWS_PER_BLOCK + wid * TILE_ROWS;
  if (row0 >= B) return;
  float* lds = smem + wid * TILE_ROWS * LDS_STRIDE;

  const int l16 = lane & 15;
  const int hi = lane >> 4;  // 0: lanes 0-15, 1: lanes 16-31
  const int T = num_steps[0];
  if (T <= 0) return;

  // ---- Preload W_ih.T / W_hh.T (32x96) as WMMA B operands in registers ----
  v2f wih[6][8], whh[6][8];
#pragma unroll
  for (int n0 = 0; n0 < 6; ++n0) {
    const int n = n0 * 16 + l16;
#pragma unroll
    for (int k = 0; k < 8; ++k) {
      const int off = n * GRU_H + 4 * k + 2 * hi;
      wih[n0][k] = *reinterpret_cast<const v2f*>(W_ih + off);
      whh[n0][k] = *reinterpret_cast<const v2f*>(W_hh + off);
    }
  }
  // Biases, folded into gate math (accumulators are zero-seeded).
  // brz[n0] = b_ih[n0 tile] + b_hh[n0 tile] for the 4 r/z tiles;
  // bin/bhn = separate n-gate biases.
  float brz[4], bin[2], bhn[2];
#pragma unroll
  for (int n0 = 0; n0 < 4; ++n0)
    brz[n0] = b_ih[n0 * 16 + l16] + b_hh[n0 * 16 + l16];
#pragma unroll
  for (int n0 = 0; n0 < 2; ++n0) {
    bin[n0] = b_ih[(4 + n0) * 16 + l16];
    bhn[n0] = b_hh[(4 + n0) * 16 + l16];
  }
  float wo[16];
#pragma unroll
  for (int j = 0; j < 16; ++j) wo[j] = W_out[hi * 16 + j];
  const float bo = b_out[0];

  // ---- A operands (batch across lanes, hidden across VGPR chunks) ----
  v2f aX[8], aH[8];
#pragma unroll
  for (int k = 0; k < 8; ++k) {
    const int off = (row0 + l16) * GRU_H + 4 * k + 2 * hi;
    aX[k] = *reinterpret_cast<const v2f*>(in_data + off);
    aH[k] = *reinterpret_cast<const v2f*>(hidden + off);
  }
  // D-layout h (batch across VGPRs, hidden across lanes) for the z*h term.
  v8f hD[2];
#pragma unroll
  for (int n0 = 0; n0 < 2; ++n0)
#pragma unroll
    for (int r = 0; r < 8; ++r)
      hD[n0][r] = hidden[(row0 + r + 8 * hi) * GRU_H + n0 * 16 + l16];

  // Shared epilogue. pr/pz/pin/phn are bias-free gate pre-activations.
  auto epilogue = [&](const v8f* pr, const v8f* pz, const v8f* pin,
                      const v8f* phn, int t) {
#pragma unroll
    for (int n0 = 0; n0 < 2; ++n0) {
#pragma unroll
      for (int r = 0; r < 8; ++r) {
        const float rg = fast_sigmoid(pr[n0][r] + brz[n0]);
        const float zg = fast_sigmoid(pz[n0][r] + brz[2 + n0]);
        const float ng =
            fast_tanh(pin[n0][r] + bin[n0] + rg * (phn[n0][r] + bhn[n0]));
        hD[n0][r] = ng + zg * (hD[n0][r] - ng);  // (1-z)*n + z*h
      }
    }
    // h_new tile (row-major [batch][hidden]) -> wave-private LDS
#pragma unroll
    for (int n0 = 0; n0 < 2; ++n0)
#pragma unroll
      for (int r = 0; r < 8; ++r)
        lds[(r + 8 * hi) * LDS_STRIDE + n0 * 16 + l16] = hD[n0][r];
    wave_lds_sync();
    // transpose back: next step's A operand
#pragma unroll
    for (int k = 0; k < 8; ++k) {
      const float* p = lds + l16 * LDS_STRIDE + 4 * k + 2 * hi;
      v2f a;
      a[0] = p[0];
      a[1] = p[1];
      aH[k] = a;
    }
    // y head: split K=32 dot across lane halves
    float yv = 0.0f;
#pragma unroll
    for (int j = 0; j < 16; ++j)
      yv += lds[l16 * LDS_STRIDE + hi * 16 + j] * wo[j];
    yv += __shfl_xor(yv, 16, 32);
    if (lane < 16) out[(row0 + l16) * T + t] = yv + bo;
    wave_lds_sync();  // keep this step's LDS reads ahead of next step's stores
  };

  // ---------------- step 0: generic (x = in_data != h) ----------------
  {
    v8f gi[6], gh[6];
#pragma unroll
    for (int n0 = 0; n0 < 6; ++n0) {
      v8f ci = {};  // folds to inline src2=0 in the first WMMA of the chain
      v8f ch = {};
#pragma unroll
      for (int k = 0; k < 8; ++k) {
        ci = wmma4(aX[k], wih[n0][k], ci);
        ch = wmma4(aH[k], whh[n0][k], ch);
      }
      gi[n0] = ci;
      gh[n0] = ch;
    }
    v8f pr[2], pz[2], pin[2], phn[2];
#pragma unroll
    for (int n0 = 0; n0 < 2; ++n0) {
      pr[n0] = gi[n0] + gh[n0];
      pz[n0] = gi[2 + n0] + gh[2 + n0];
      pin[n0] = gi[4 + n0];
      phn[n0] = gh[4 + n0];
    }
    epilogue(pr, pz, pin, phn, 0);
  }

  // ---- fuse r/z weights in place: W_rz = W_ih_rz + W_hh_rz (x==h, t>=1) ----
#pragma unroll
  for (int n0 = 0; n0 < 4; ++n0)
#pragma unroll
    for (int k = 0; k < 8; ++k) wih[n0][k] += whh[n0][k];

  // ---------------- steps 1..T-1: fused path (64 WMMAs/step) ----------------
  for (int t = 1; t < T; ++t) {
    v8f grz[4], gin[2], ghn[2];
#pragma unroll
    for (int n0 = 0; n0 < 4; ++n0) {
      v8f c = {};
#pragma unroll
      for (int k = 0; k < 8; ++k) c = wmma4(aH[k], wih[n0][k], c);
      grz[n0] = c;
    }
#pragma unroll
    for (int n0 = 0; n0 < 2; ++n0) {
      v8f ci = {};
      v8f ch = {};
#pragma unroll
      for (int k = 0; k < 8; ++k) {
        ci = wmma4(aH[k], wih[4 + n0][k], ci);
        ch = wmma4(aH[k], whh[4 + n0][k], ch);
      }
      gin[n0] = ci;
      ghn[n0] = ch;
    }
    epilogue(&grz[0], &grz[2], gin, ghn, t);
  }
}

extern "C" void kernel_launch(void* const* d_in, const int* in_sizes, int n_in,
                              void* d_out, int out_size, void* d_ws,
                              size_t ws_size, hipStream_t stream) {
  const float* in_data = (const float*)d_in[0];
  const float* hidden = (const float*)d_in[1];
  const float* W_ih = (const float*)d_in[2];
  const float* W_hh = (const float*)d_in[3];
  const float* b_ih = (const float*)d_in[4];
  const float* b_hh = (const float*)d_in[5];
  const float* W_out = (const float*)d_in[6];
  const float* b_out = (const float*)d_in[7];
  const int* num_steps = (const int*)d_in[8];
  float* out = (float*)d_out;

  const int B = in_sizes[0] / GRU_H;  // in_data is [B, 32]
  const int grid = (B + ROWS_PER_BLOCK - 1) / ROWS_PER_BLOCK;

  GRU_DE_44968307589653_kernel<<<grid, 256, 0, stream>>>(
      in_data, hidden, W_ih, W_hh, b_ih, b_hh, W_out, b_out, num_steps, out, B);
}